// SoftSimpleShader_77979426226426
// MI455X (gfx1250) — compile-verified
//
#include <hip/hip_runtime.h>
#include <stdint.h>

// --- blending constants (from reference) ---
#define SIGMA_INV 1.0e4f          // 1/1e-4
#define GAMMA_INV 1.0e4f          // 1/1e-4
#define ZNEAR 0.1f
#define ZFAR  50.0f
#define EPSV  1e-10f
#define ZRANGE_INV (1.0f / (ZFAR - ZNEAR))

#define KFACES 8
#define TILE 256                                   // pixels per tile == blockDim.x
#define TEXPP (KFACES * 3)                         // 24 floats per pixel
#define TILE_TEX_FLOATS (TILE * TEXPP)             // 6144 floats = 24 KB
#define TILE_TEX_E8 (TILE_TEX_FLOATS / 2)          // 3072 8-byte elements

// TDM descriptor group vector types (clang-23 / therock-10.0: 6-arg builtin)
typedef unsigned int u32x4 __attribute__((ext_vector_type(4)));
typedef int          i32x8 __attribute__((ext_vector_type(8)));
typedef int          i32x4 __attribute__((ext_vector_type(4)));
// native clang vectors for nontemporal 128-bit accesses
typedef float        f32x4 __attribute__((ext_vector_type(4)));

__global__ __launch_bounds__(TILE) void softmax_blend_kernel(
    const float* __restrict__ texels,
    const float* __restrict__ dists,
    const float* __restrict__ zbuf,
    const int*   __restrict__ p2f,
    float* __restrict__ out,
    int P, int ntiles)
{
    // double-buffered texel staging, filled by the Tensor Data Mover
    __shared__ __attribute__((aligned(16))) float lds[2 * TILE_TEX_FLOATS]; // 48 KB

    const int tid = threadIdx.x;
    const bool wave0 = (tid < 32);

    // low 32 bits of a generic shared pointer == LDS byte offset
    const uint32_t lds_base = (uint32_t)(uintptr_t)(void*)&lds[0];
    const unsigned long long tex_base = (unsigned long long)(uintptr_t)texels;
    const unsigned long long total_e8 =
        ((unsigned long long)P * (unsigned long long)TEXPP) >> 1; // 8B elements

    // Issue one TDM DMA: 24 KB contiguous texel tile -> LDS buffer `buf`
    auto stage = [&](int tile, int buf) {
        unsigned long long start_e8 = (unsigned long long)tile * TILE_TEX_E8;
        unsigned long long gaddr = tex_base + start_e8 * 8ull;
        uint32_t laddr = lds_base + (uint32_t)buf * (TILE_TEX_FLOATS * 4);
        unsigned long long rem = total_e8 - start_e8;
        uint32_t tdim0 = (rem > 0xFFFFFFFFull) ? 0xFFFFFFFFu : (uint32_t)rem;

        u32x4 g0;
        g0.x = 1u;                                            // count=1, no gather
        g0.y = laddr;                                         // lds_addr (bytes)
        g0.z = (uint32_t)(gaddr & 0xFFFFFFFFull);             // global_addr[31:0]
        g0.w = (uint32_t)((gaddr >> 32) & 0x1FFFFFFull)       // global_addr[56:32]
               | (2u << 30);                                  // type = 2 ("image")
        i32x8 g1;
        g1[0] = (int)(3u << 16);                              // data_size = 8 bytes
        g1[1] = (int)((tdim0 & 0xFFFFu) << 16);               // tensor_dim0[15:0]
        g1[2] = (int)((tdim0 >> 16) & 0xFFFFu);               // tensor_dim0[31:16], tensor_dim1=0
        g1[3] = (int)((uint32_t)TILE_TEX_E8 << 16);           // tile_dim0 = 3072 (1-D tile)
        g1[4] = 0;                                            // tile_dim1/2 unused
        g1[5] = (int)TILE_TEX_E8;                             // tensor_dim0_stride (unused, 1-D)
        g1[6] = 0;
        g1[7] = 0;
        i32x4 gz4 = {0, 0, 0, 0};
        i32x8 gz8 = {0, 0, 0, 0, 0, 0, 0, 0};
        __builtin_amdgcn_tensor_load_to_lds(g0, g1, gz4, gz4, gz8, 0);
    };

    int t = blockIdx.x;
    const int stride = gridDim.x;
    int parity = 0;

    if (t < ntiles && wave0) stage(t, 0);                     // prime the pipeline

    for (; t < ntiles; t += stride) {
        if (wave0) __builtin_amdgcn_s_wait_tensorcnt(0);      // buf[parity] landed
        __syncthreads();                                      // make LDS visible to all waves

        int tn = t + stride;
        if (tn < ntiles && wave0) stage(tn, parity ^ 1);      // overlap next DMA with compute

        int p = t * TILE + tid;
        if (p < P) {
            // coalesced 128-bit global loads, non-temporal (streamed once, keep L2 clean)
            const f32x4* dp = (const f32x4*)(dists + (size_t)p * KFACES);
            const f32x4* zp = (const f32x4*)(zbuf  + (size_t)p * KFACES);
            const i32x4* fp = (const i32x4*)(p2f   + (size_t)p * KFACES);
            f32x4 d0 = __builtin_nontemporal_load(dp);
            f32x4 d1 = __builtin_nontemporal_load(dp + 1);
            f32x4 z0 = __builtin_nontemporal_load(zp);
            f32x4 z1 = __builtin_nontemporal_load(zp + 1);
            i32x4 f0 = __builtin_nontemporal_load(fp);
            i32x4 f1 = __builtin_nontemporal_load(fp + 1);

            float dk[KFACES] = {d0.x, d0.y, d0.z, d0.w, d1.x, d1.y, d1.z, d1.w};
            float zk[KFACES] = {z0.x, z0.y, z0.z, z0.w, z1.x, z1.y, z1.z, z1.w};
            int   fk[KFACES] = {f0.x, f0.y, f0.z, f0.w, f1.x, f1.y, f1.z, f1.w};

            float prob[KFACES], zinv[KFACES];
            float alpha = 1.0f, zmax = EPSV;
#pragma unroll
            for (int k = 0; k < KFACES; ++k) {
                float m  = (fk[k] >= 0) ? 1.0f : 0.0f;
                float pr = m / (1.0f + __expf(dk[k] * SIGMA_INV)); // sigmoid(-d/sigma)*mask
                prob[k] = pr;
                alpha *= (1.0f - pr);
                float zi = (ZFAR - zk[k]) * ZRANGE_INV * m;
                zinv[k] = zi;
                zmax = fmaxf(zmax, zi);
            }

            // texels from the TDM-staged LDS buffer (24 contiguous floats per lane)
            float tex[TEXPP];
            const float* src = &lds[parity * TILE_TEX_FLOATS + tid * TEXPP];
#pragma unroll
            for (int j = 0; j < TEXPP; ++j) tex[j] = src[j];

            float wsum = 0.0f, r = 0.0f, g = 0.0f, b = 0.0f;
#pragma unroll
            for (int k = 0; k < KFACES; ++k) {
                float w = prob[k] * __expf((zinv[k] - zmax) * GAMMA_INV);
                wsum += w;
                r += w * tex[k * 3 + 0];
                g += w * tex[k * 3 + 1];
                b += w * tex[k * 3 + 2];
            }

            float delta = fmaxf(__expf((EPSV - zmax) * GAMMA_INV), EPSV);
            float inv = 1.0f / (wsum + delta);
            f32x4 o;
            o.x = (r + delta) * inv;   // background color is (1,1,1)
            o.y = (g + delta) * inv;
            o.z = (b + delta) * inv;
            o.w = 1.0f - alpha;
            __builtin_nontemporal_store(o, (f32x4*)out + p);
        }

        __syncthreads();               // all reads of buf[parity] done before reuse
        parity ^= 1;
    }
}

extern "C" void kernel_launch(void* const* d_in, const int* in_sizes, int n_in,
                              void* d_out, int out_size, void* d_ws, size_t ws_size,
                              hipStream_t stream) {
    const float* texels = (const float*)d_in[0];
    const float* dists  = (const float*)d_in[1];
    const float* zbuf   = (const float*)d_in[2];
    const int*   p2f    = (const int*)d_in[3];
    float* out = (float*)d_out;

    int P = in_sizes[1] / KFACES;                 // N*H*W pixels
    int ntiles = (P + TILE - 1) / TILE;
    int blocks = ntiles < 2048 ? ntiles : 2048;   // persistent blocks -> DMA pipelining
    if (blocks < 1) blocks = 1;

    softmax_blend_kernel<<<blocks, TILE, 0, stream>>>(texels, dists, zbuf, p2f, out, P, ntiles);
}